// DynamicPointConvBackBone_71184787964124
// MI455X (gfx1250) — compile-verified
//
#include <hip/hip_runtime.h>
#include <hip/hip_bf16.h>

typedef float v2f __attribute__((ext_vector_type(2)));
typedef float v8f __attribute__((ext_vector_type(8)));
typedef int   v4i __attribute__((ext_vector_type(4)));

// Fixed problem geometry (from the reference's setup_inputs); runtime dims
// (N, M, B, K, totalRows) are derived on the host and passed in.
#define K3    27
#define CIN   64
#define KTOT  1728          // K3*CIN
#define COUT  128
#define LDA   1732          // KTOT + 4 dword pad -> conflict-free A-frag reads
#define LDO   132           // COUT + 4 pad
#define NCHUNK 432          // KTOT / 4
#define EPS   1e-3f

#if __has_builtin(__builtin_amdgcn_global_load_async_to_lds_b128)
#define USE_ASYNC_GATHER 1
#else
#define USE_ASYNC_GATHER 0
#endif

// Repack W ([KTOT,COUT] row-major) into row-pair-interleaved float2:
// Wp[q*COUT + c] = { W[2q][c], W[2q+1][c] },  q = 0..KTOT/2-1
// -> each WMMA B-fragment becomes ONE global_load_b64, contiguous per half-wave.
__global__ __launch_bounds__(256) void pack_w_kernel(
    const float* __restrict__ Wm, float2* __restrict__ Wp, int total)
{
    int e = blockIdx.x * 256 + threadIdx.x;
    if (e < total) {
        int q = e / COUT;
        int c = e - q * COUT;
        Wp[e] = make_float2(Wm[(size_t)(2 * q) * COUT + c],
                            Wm[(size_t)(2 * q + 1) * COUT + c]);
    }
}

template <bool PACKED>
__global__ __launch_bounds__(256) void dpc_wmma_kernel(
    const float* __restrict__ features,     // [N, CIN]
    const float* __restrict__ center,       // [M, 3]
    const int*   __restrict__ vidx,         // [M, K3]
    const int*   __restrict__ numlist,      // [B]
    const float* __restrict__ Wany,         // PACKED ? float2-packed : [KTOT, COUT]
    const float* __restrict__ gamma,        // [COUT]
    const float* __restrict__ beta,         // [COUT]
    float* __restrict__ out,                // features [totalRows,COUT] ++ coords [totalRows,4]
    int N, int M, int B, int K, int totalRows)
{
    __shared__ float ldsA[16 * LDA];        // gathered A panel, 16 rows x KTOT
    __shared__ float ldsOut[16 * LDO];      // 16 x COUT result tile
    __shared__ int   sSrc[16];
    __shared__ int   sValid[16];

    const int tid     = threadIdx.x;
    const int rowBase = blockIdx.x * 16;

    // ---- per-row source mapping (resample inverse) ----
    if (tid < 16) {
        int o = rowBase + tid;
        int src = 0, vld = 0;
        if (o < totalRows) {
            int b = o / K;
            int j = o - b * K;
            int off = 0;
            for (int i = 0; i < b; ++i) off += numlist[i];
            int end = numlist[b]; if (end > K) end = K;
            vld = (j < end) ? 1 : 0;
            src = off + j;
            if (src < 0) src = 0;
            if (src > M - 1) src = M - 1;
        }
        sSrc[tid]   = src;
        sValid[tid] = vld;
    }
    __syncthreads();

    // ---- point_coords output (totalRows x 4 after the feature block) ----
    if (tid < 64) {
        int m = tid >> 2, c = tid & 3;
        int o = rowBase + m;
        if (o < totalRows) {
            float val;
            int b = o / K;
            if (c == 0) {
                val = (b < B - 1) ? (float)(b + 1) : 0.0f;
            } else {
                val = sValid[m] ? center[(size_t)sSrc[m] * 3 + (c - 1)] : 0.0f;
            }
            float* outC = out + (size_t)totalRows * COUT;
            outC[(size_t)o * 4 + c] = val;
        }
    }

    // ---- gather: 16 rows x (K3*CIN) into LDS; 16 threads per (m,k) pair ----
    {
        const int pairSlot = tid >> 4;        // 0..15
        const int c        = (tid & 15) * 4;  // channel base (float4)
        for (int p = pairSlot; p < 16 * K3; p += 16) {
            int m = p / K3;
            int k = p - m * K3;
            float* ldst = ldsA + m * LDA + k * CIN + c;
            int vld = sValid[m];
            int ii  = -1;
            if (vld) ii = vidx[(size_t)sSrc[m] * K3 + k];
            if (vld && ii >= 0) {
                if (ii > N - 1) ii = N - 1;
                const float* g = features + (size_t)ii * CIN + c;
#if USE_ASYNC_GATHER
                __builtin_amdgcn_global_load_async_to_lds_b128(
                    (__attribute__((address_space(1))) v4i*)g,
                    (__attribute__((address_space(3))) v4i*)ldst,
                    0, 0);
#else
                *(float4*)ldst = *(const float4*)g;
#endif
            } else {
                *(float4*)ldst = make_float4(0.f, 0.f, 0.f, 0.f);
            }
        }
    }
#if USE_ASYNC_GATHER
#if __has_builtin(__builtin_amdgcn_s_wait_asynccnt)
    __builtin_amdgcn_s_wait_asynccnt(0);
#else
    asm volatile("s_wait_asynccnt 0x0" ::: "memory");
#endif
#endif
    __syncthreads();

    // ---- WMMA main loop: each wave owns a 16x16 tile of the 16x128 output ----
    const int lane = tid & 31;
    const int wv   = tid >> 5;               // 0..7 -> column tile
    const int half = lane >> 4;              // K sub-position within fragment
    const int mrow = lane & 15;
    const int col  = wv * 16 + (lane & 15);

    const float2* Arow2 = (const float2*)(ldsA + mrow * LDA + 2 * half); // 8B aligned
    const float*  Wcol  = Wany + (size_t)(2 * half) * COUT + col;        // direct path
    const float2* Bpk   = ((const float2*)Wany) + (size_t)half * COUT + col; // packed path

    v8f acc0 = {0.f,0.f,0.f,0.f,0.f,0.f,0.f,0.f};
    v8f acc1 = {0.f,0.f,0.f,0.f,0.f,0.f,0.f,0.f};

#pragma unroll 4
    for (int kk = 0; kk < NCHUNK; kk += 2) {
        v2f a0, a1, b0, b1;
        float2 t0 = Arow2[2 * kk];
        float2 t1 = Arow2[2 * kk + 2];
        a0.x = t0.x; a0.y = t0.y;
        a1.x = t1.x; a1.y = t1.y;
        if (PACKED) {
            float2 u0 = Bpk[(size_t)(2 * kk) * COUT];      // {W[4kk+2h][col], W[4kk+2h+1][col]}
            float2 u1 = Bpk[(size_t)(2 * kk + 2) * COUT];
            b0.x = u0.x; b0.y = u0.y;
            b1.x = u1.x; b1.y = u1.y;
        } else {
            b0.x = Wcol[(size_t)(4 * kk)     * COUT];
            b0.y = Wcol[(size_t)(4 * kk + 1) * COUT];
            b1.x = Wcol[(size_t)(4 * kk + 4) * COUT];
            b1.y = Wcol[(size_t)(4 * kk + 5) * COUT];
        }
        acc0 = __builtin_amdgcn_wmma_f32_16x16x4_f32(
                   false, a0, false, b0, (short)0, acc0, false, false);
        acc1 = __builtin_amdgcn_wmma_f32_16x16x4_f32(
                   false, a1, false, b1, (short)0, acc1, false, false);
    }
    acc0 = acc0 + acc1;

    // ---- spill tile to LDS (row = r + 8*half, col) ----
    {
        const int rbase = half * 8;
#pragma unroll
        for (int r = 0; r < 8; ++r)
            ldsOut[(rbase + r) * LDO + col] = acc0[r];
    }
    __syncthreads();

    // ---- LayerNorm + affine + ReLU: each wave handles 2 rows ----
#pragma unroll
    for (int rr = 0; rr < 2; ++rr) {
        int R = wv * 2 + rr;
        int o = rowBase + R;
        if (o >= totalRows) continue;
        const float* orow = ldsOut + R * LDO;
        float4 v = *(const float4*)(orow + lane * 4);
        float s  = v.x + v.y + v.z + v.w;
        float s2 = v.x * v.x + v.y * v.y + v.z * v.z + v.w * v.w;
#pragma unroll
        for (int off = 16; off >= 1; off >>= 1) {
            s  += __shfl_xor(s,  off, 32);
            s2 += __shfl_xor(s2, off, 32);
        }
        float4 y = make_float4(0.f, 0.f, 0.f, 0.f);
        if (sValid[R]) {
            float mean = s * (1.0f / COUT);
            float var  = s2 * (1.0f / COUT) - mean * mean;
            float rstd = rsqrtf(var + EPS);
            float4 g  = *(const float4*)(gamma + lane * 4);
            float4 bt = *(const float4*)(beta  + lane * 4);
            y.x = fmaxf((v.x - mean) * rstd * g.x + bt.x, 0.f);
            y.y = fmaxf((v.y - mean) * rstd * g.y + bt.y, 0.f);
            y.z = fmaxf((v.z - mean) * rstd * g.z + bt.z, 0.f);
            y.w = fmaxf((v.w - mean) * rstd * g.w + bt.w, 0.f);
        }
        *(float4*)(out + (size_t)o * COUT + lane * 4) = y;
    }
}

extern "C" void kernel_launch(void* const* d_in, const int* in_sizes, int n_in,
                              void* d_out, int out_size, void* d_ws, size_t ws_size,
                              hipStream_t stream) {
    const float* features = (const float*)d_in[0];
    const float* center   = (const float*)d_in[1];
    const int*   vidx     = (const int*)d_in[2];
    const int*   numlist  = (const int*)d_in[3];
    const float* Wm       = (const float*)d_in[4];
    const float* gamma    = (const float*)d_in[5];
    const float* beta     = (const float*)d_in[6];
    // num_keypoints (d_in[7]) lives on device; derive K from sizes instead.

    int M     = in_sizes[1] / 3;                    // 40000
    int k3    = in_sizes[2] / M;                    // 27
    int cOut  = in_sizes[5];                        // 128
    int cIn   = in_sizes[4] / (k3 * cOut);          // 64
    int N     = in_sizes[0] / cIn;                  // 200000
    int B     = in_sizes[3];                        // 4
    int K     = out_size / (B * (cOut + 4));        // 2048
    int totalRows = B * K;                          // 8192

    int grid = (totalRows + 15) / 16;               // 512

    const size_t needW = (size_t)KTOT * COUT * sizeof(float); // 884,736 B
    if (ws_size >= needW) {
        float2* Wp = (float2*)d_ws;
        int total  = (KTOT / 2) * COUT;             // 110,592 float2 elements
        pack_w_kernel<<<(total + 255) / 256, 256, 0, stream>>>(Wm, Wp, total);
        dpc_wmma_kernel<true><<<grid, 256, 0, stream>>>(
            features, center, vidx, numlist, (const float*)Wp, gamma, beta,
            (float*)d_out, N, M, B, K, totalRows);
    } else {
        dpc_wmma_kernel<false><<<grid, 256, 0, stream>>>(
            features, center, vidx, numlist, Wm, gamma, beta,
            (float*)d_out, N, M, B, K, totalRows);
    }
}